// FooNet_57526791962848
// MI455X (gfx1250) — compile-verified
//
#include <hip/hip_runtime.h>
#include <hip/hip_bf16.h>

// ---------------------------------------------------------------------------
// HolE-style scoring, fused for gfx1250 (MI455X), wave32.
// Each lane owns ONE element (32 distinct elements / wave / iteration = two
// 16-wide WMMA tiles). Per element:
//   front-end (VALU): gather e1,r,e2 (D=10), circular correlation,
//                     x = r*cc, xb = x - 0.5*b
//   tiles A,B (WMMA f32 16x16x4, K padded 10->12):
//                     h = xb @ V,  t = Wsym @ h   (V = inv(W+W^T - I))
//   epilogue: score = h.t + h.b - |h-x|^2  (xor16 reduce, 1 store/lane)
// Cross-half data movement: xor-16 shuffles only; all vector indices are
// compile-time constants (single cndmask on `half`, no select ladders).
// ---------------------------------------------------------------------------

typedef __attribute__((ext_vector_type(2))) float v2f;
typedef __attribute__((ext_vector_type(8))) float v8f;

#define DD     10
#define TOTAL  (8192 * 256)       // 2,097,152 scored elements
#define NPAIRS (TOTAL / 32)       // 65,536 tile-pairs (2 WMMA tiles each)

// --------------------------- setup kernel ----------------------------------
// One thread, once: Wsym = W + W^T ; V = inv(Wsym - I) (Gauss-Jordan, partial
// pivoting, matches jnp.linalg.inv in spirit). Store zero-padded A operands:
//   ws[  0..191] : A1[m][k] = V[k][m]   (V^T), 16x12, zero padded
//   ws[192..383] : A2[m][k] = Wsym[m][k],      16x12, zero padded
__global__ void hole_prep_kernel(const float* __restrict__ W,
                                 float* __restrict__ ws) {
  if (blockIdx.x != 0 || threadIdx.x != 0) return;

  float Ws[DD][DD], M[DD][DD], V[DD][DD];
  for (int i = 0; i < DD; ++i)
    for (int j = 0; j < DD; ++j)
      Ws[i][j] = W[i * DD + j] + W[j * DD + i];

  for (int i = 0; i < DD; ++i)
    for (int j = 0; j < DD; ++j) {
      M[i][j] = Ws[i][j] - (i == j ? 1.0f : 0.0f);   // LAM = 1.0
      V[i][j] = (i == j) ? 1.0f : 0.0f;
    }

  for (int c = 0; c < DD; ++c) {
    int p = c;
    float best = fabsf(M[c][c]);
    for (int r = c + 1; r < DD; ++r) {
      float a = fabsf(M[r][c]);
      if (a > best) { best = a; p = r; }
    }
    if (p != c) {
      for (int j = 0; j < DD; ++j) {
        float tm = M[c][j]; M[c][j] = M[p][j]; M[p][j] = tm;
        float tv = V[c][j]; V[c][j] = V[p][j]; V[p][j] = tv;
      }
    }
    float inv = 1.0f / M[c][c];
    for (int j = 0; j < DD; ++j) { M[c][j] *= inv; V[c][j] *= inv; }
    for (int r = 0; r < DD; ++r) {
      if (r == c) continue;
      float f = M[r][c];
      for (int j = 0; j < DD; ++j) {
        M[r][j] -= f * M[c][j];
        V[r][j] -= f * V[c][j];
      }
    }
  }

  for (int m = 0; m < 16; ++m)
    for (int k = 0; k < 12; ++k) {
      bool in = (m < DD) && (k < DD);
      ws[m * 12 + k]       = in ? V[k][m]  : 0.0f;   // A1 = V^T
      ws[192 + m * 12 + k] = in ? Ws[m][k] : 0.0f;   // A2 = Wsym
    }
}

// ---------------------------- main kernel ----------------------------------
__global__ __launch_bounds__(256) void hole_score_kernel(
    const int* __restrict__ samples,   // [B,N,3] int32
    const float* __restrict__ ent,     // [NE,10]
    const float* __restrict__ rel,     // [NR,10]
    const float* __restrict__ bvec,    // [10]
    const float* __restrict__ ws,      // A1pad(16x12), A2pad(16x12)
    float* __restrict__ out,           // [B*N]
    int nWaves) {
  const int lane = (int)(threadIdx.x & 31);
  const int half = lane >> 4;          // 0: lanes 0-15, 1: lanes 16-31
  const int mrow = lane & 15;
  const int wave = (int)((blockIdx.x * blockDim.x + threadIdx.x) >> 5);

  // Constant A operands in native A layout:
  // VGPR v of chunk c supplies A[M = mrow][K = 4c + 2*half + v].
  const float* A1 = ws;
  const float* A2 = ws + 192;
  v2f a1[3], a2[3];
#pragma unroll
  for (int c = 0; c < 3; ++c) {
    const int idx = mrow * 12 + 4 * c + 2 * half;
    a1[c].x = A1[idx]; a1[c].y = A1[idx + 1];
    a2[c].x = A2[idx]; a2[c].y = A2[idx + 1];
  }

  float bb[DD];
#pragma unroll
  for (int k = 0; k < DD; ++k) bb[k] = bvec[k];

  for (int p = wave; p < NPAIRS; p += nWaves) {  // uniform per wave: EXEC all-1
    const int base = p * 32;
    const int e    = base + lane;                // this lane's OWN element

    const int* sp = samples + 3 * e;
    const int s0 = sp[0], s1 = sp[1], s2 = sp[2];

    // Rows are 40B (8B-aligned) -> five b64 loads each.
    float e1[DD], rr[DD], e2[DD];
    const float2* p1 = reinterpret_cast<const float2*>(ent + DD * s0);
    const float2* pr = reinterpret_cast<const float2*>(rel + DD * s1);
    const float2* p2 = reinterpret_cast<const float2*>(ent + DD * s2);
#pragma unroll
    for (int i = 0; i < 5; ++i) {
      float2 v1 = p1[i]; e1[2 * i] = v1.x; e1[2 * i + 1] = v1.y;
      float2 vr = pr[i]; rr[2 * i] = vr.x; rr[2 * i + 1] = vr.y;
      float2 v2 = p2[i]; e2[2 * i] = v2.x; e2[2 * i + 1] = v2.y;
    }

    // circular correlation + harmonic embedding (own element only)
    float x[DD], xb[DD];
#pragma unroll
    for (int k = 0; k < DD; ++k) {
      float acc = 0.0f;
#pragma unroll
      for (int j = 0; j < DD; ++j)
        acc = __builtin_fmaf(e1[j], e2[(j + k) % DD], acc);
      x[k]  = rr[k] * acc;
      xb[k] = x[k] - 0.5f * bb[k];
    }

    // partner's xb (element lane^16); x reconstructs as xs[k] + 0.5*b[k]
    float xs[DD];
#pragma unroll
    for (int k = 0; k < DD; ++k) xs[k] = __shfl_xor(xb[k], 16, 32);

    // ---- stage 1: h^T = V^T * xb^T, two 16-element tiles -----------------
    // B layout: VGPR v, lane-half hh supplies B[K = 4c + 2*hh + v][N = mrow].
    // Tile A: columns owned by lanes 0-15; tile B: by lanes 16-31.
    v8f hA = {}, hB = {};
    {
      v2f B; B.x = half ? xs[2] : xb[0]; B.y = half ? xs[3] : xb[1];
      hA = __builtin_amdgcn_wmma_f32_16x16x4_f32(false, a1[0], false, B, (short)0, hA, false, false);
      B.x = half ? xs[6] : xb[4]; B.y = half ? xs[7] : xb[5];
      hA = __builtin_amdgcn_wmma_f32_16x16x4_f32(false, a1[1], false, B, (short)0, hA, false, false);
      B.x = half ? 0.0f : xb[8]; B.y = half ? 0.0f : xb[9];
      hA = __builtin_amdgcn_wmma_f32_16x16x4_f32(false, a1[2], false, B, (short)0, hA, false, false);

      B.x = half ? xb[2] : xs[0]; B.y = half ? xb[3] : xs[1];
      hB = __builtin_amdgcn_wmma_f32_16x16x4_f32(false, a1[0], false, B, (short)0, hB, false, false);
      B.x = half ? xb[6] : xs[4]; B.y = half ? xb[7] : xs[5];
      hB = __builtin_amdgcn_wmma_f32_16x16x4_f32(false, a1[1], false, B, (short)0, hB, false, false);
      B.x = half ? 0.0f : xs[8]; B.y = half ? 0.0f : xs[9];
      hB = __builtin_amdgcn_wmma_f32_16x16x4_f32(false, a1[2], false, B, (short)0, hB, false, false);
    }
    // D layout: half 0 holds dims 0..7, half 1 dims 8..15 (>=10 exact zeros).

    // ---- stage 2: t^T = Wsym * h^T (per tile; 6 xor16 shuffles each) -----
    v8f tA = {}, tB = {};
    {
      const float Q0 = __shfl_xor(hA[0], 16, 32), Q1 = __shfl_xor(hA[1], 16, 32);
      const float Q2 = __shfl_xor(hA[2], 16, 32), Q3 = __shfl_xor(hA[3], 16, 32);
      const float Q6 = __shfl_xor(hA[6], 16, 32), Q7 = __shfl_xor(hA[7], 16, 32);
      v2f B; B.x = half ? Q2 : hA[0]; B.y = half ? Q3 : hA[1];
      tA = __builtin_amdgcn_wmma_f32_16x16x4_f32(false, a2[0], false, B, (short)0, tA, false, false);
      B.x = half ? Q6 : hA[4]; B.y = half ? Q7 : hA[5];
      tA = __builtin_amdgcn_wmma_f32_16x16x4_f32(false, a2[1], false, B, (short)0, tA, false, false);
      B.x = half ? 0.0f : Q0; B.y = half ? 0.0f : Q1;
      tA = __builtin_amdgcn_wmma_f32_16x16x4_f32(false, a2[2], false, B, (short)0, tA, false, false);
    }
    {
      const float Q0 = __shfl_xor(hB[0], 16, 32), Q1 = __shfl_xor(hB[1], 16, 32);
      const float Q2 = __shfl_xor(hB[2], 16, 32), Q3 = __shfl_xor(hB[3], 16, 32);
      const float Q6 = __shfl_xor(hB[6], 16, 32), Q7 = __shfl_xor(hB[7], 16, 32);
      v2f B; B.x = half ? Q2 : hB[0]; B.y = half ? Q3 : hB[1];
      tB = __builtin_amdgcn_wmma_f32_16x16x4_f32(false, a2[0], false, B, (short)0, tB, false, false);
      B.x = half ? Q6 : hB[4]; B.y = half ? Q7 : hB[5];
      tB = __builtin_amdgcn_wmma_f32_16x16x4_f32(false, a2[1], false, B, (short)0, tB, false, false);
      B.x = half ? 0.0f : Q0; B.y = half ? 0.0f : Q1;
      tB = __builtin_amdgcn_wmma_f32_16x16x4_f32(false, a2[2], false, B, (short)0, tB, false, false);
    }

    // ---- stage 3: score = h.t + h.b - |h - x|^2 --------------------------
    // Tile A element for this lane: half0 -> own, half1 -> partner.
    float qA = 0.0f, lA = 0.0f, sA = 0.0f;
    float qB = 0.0f, lB = 0.0f, sB = 0.0f;
#pragma unroll
    for (int r = 0; r < 8; ++r) {
      const float bv = half ? (r < 2 ? bb[r + 8] : 0.0f) : bb[r];

      const float hvA = hA[r];
      qA = __builtin_fmaf(hvA, tA[r], qA);
      lA = __builtin_fmaf(hvA, bv, lA);
      const float xvA = half ? (r < 2 ? __builtin_fmaf(0.5f, bb[r + 8], xs[r + 8]) : 0.0f)
                             : x[r];
      const float dA = hvA - xvA;
      sA = __builtin_fmaf(dA, dA, sA);

      const float hvB = hB[r];
      qB = __builtin_fmaf(hvB, tB[r], qB);
      lB = __builtin_fmaf(hvB, bv, lB);
      const float xvB = half ? (r < 2 ? x[r + 8] : 0.0f)
                             : __builtin_fmaf(0.5f, bb[r], xs[r]);
      const float dB = hvB - xvB;
      sB = __builtin_fmaf(dB, dB, sB);
    }
    float partA = qA + lA - sA;
    float partB = qB + lB - sB;
    partA += __shfl_xor(partA, 16, 32);
    partB += __shfl_xor(partB, 16, 32);

    // lane L's own element: tile A col mrow (half0) / tile B col mrow (half1)
    out[e] = half ? partB : partA;
  }
}

// ----------------------------- launcher ------------------------------------
extern "C" void kernel_launch(void* const* d_in, const int* in_sizes, int n_in,
                              void* d_out, int out_size, void* d_ws, size_t ws_size,
                              hipStream_t stream) {
  const int*   samples = (const int*)d_in[0];
  const float* ent     = (const float*)d_in[1];
  const float* rel     = (const float*)d_in[2];
  const float* W       = (const float*)d_in[3];
  const float* b       = (const float*)d_in[4];
  float* out = (float*)d_out;
  float* ws  = (float*)d_ws;

  hole_prep_kernel<<<1, 32, 0, stream>>>(W, ws);

  const int blocks  = 2048;
  const int threads = 256;                       // 8 wave32 per block
  const int nWaves  = blocks * threads / 32;     // 16384 waves, 4 pairs each
  hole_score_kernel<<<blocks, threads, 0, stream>>>(
      samples, ent, rel, b, ws, out, nWaves);
}